// MarketGraphNN_4776003633739
// MI455X (gfx1250) — compile-verified
//
#include <hip/hip_runtime.h>
#include <math.h>

// ---------------------------------------------------------------------------
// GCN 3-layer forward for MI455X (gfx1250, wave32, WMMA)
//   layer: support = h @ W   (V_WMMA_F32_16X16X4_F32, exact fp32 math,
//                             LDS-tiled weights, async global->LDS staging)
//          agg[dst] += w * support[src]   (fp32 global atomics, L2-resident)
//          h = relu(agg + b)   (final layer: log_softmax(agg + b))
// ---------------------------------------------------------------------------

typedef __attribute__((ext_vector_type(2))) float v2f;
typedef __attribute__((ext_vector_type(8))) float v8f;
typedef __attribute__((ext_vector_type(4))) int v4i;

#define GCN_NODES 30000
#define GCN_EDGES 480000
#define MTILES (GCN_NODES / 16)  // 1875 sixteen-row tiles

#define AS_GLOBAL __attribute__((address_space(1)))
#define AS_LDS __attribute__((address_space(3)))

#if defined(__has_builtin)
#if __has_builtin(__builtin_amdgcn_global_load_async_to_lds_b128) && \
    __has_builtin(__builtin_amdgcn_s_wait_asynccnt)
#define USE_ASYNC_LDS 1
#endif
#endif

// Stage 16 bytes global -> LDS. Async (ASYNCcnt) when available.
__device__ __forceinline__ void stage_b128(const float* g, float* l) {
#if USE_ASYNC_LDS
  // Builtin param types (from compiler diagnostic): global v4i*, lds v4i*,
  // imm offset, imm cpol. Pointers rebuilt via integer casts: flat->global is
  // identity; flat LDS addr truncated to 32 bits is the LDS offset (ISA 10.2).
  __builtin_amdgcn_global_load_async_to_lds_b128(
      (AS_GLOBAL v4i*)(uintptr_t)g,
      (AS_LDS v4i*)(uint32_t)(uintptr_t)l, 0, 0);
#else
  *(float4*)l = *(const float4*)g;
#endif
}

// ---------------------------------------------------------------------------
// Tiled GEMM: C[M,N] = A[M,K] @ B[K,N], fp32 WMMA 16x16x4.
// Block = 256 threads = 8 waves. Wave w computes rows [16*(8*bx+w), +16),
// all waves share columns [n0, n0+16*NSUB) whose B-tile is staged in LDS
// in KC-deep chunks, double-buffered, fetched with async global->LDS.
// ---------------------------------------------------------------------------
template <int NSUB, int KC>
__global__ __launch_bounds__(256) void gemm_wmma_tiled(
    const float* __restrict__ A, const float* __restrict__ B,
    float* __restrict__ C, int K, int N) {
  constexpr int LDN = 16 * NSUB;              // LDS row width (floats)
  constexpr int ROW_F4 = LDN / 4;             // float4s per B row
  constexpr int PASSES = (KC * LDN) / 1024;   // float4s per thread per chunk
  static_assert((KC * LDN) % 1024 == 0, "chunk must fill 256 threads x4");

  __shared__ float bsm[2][KC * LDN];

  const int tid = threadIdx.x;
  const int wave = tid >> 5;
  const int lane = tid & 31;
  const int hf = lane >> 4;  // half-wave selects K pair
  const int lm = lane & 15;  // M (A frag) / N (B,C frags)

  const int mt_raw = blockIdx.x * 8 + wave;
  const int mt = mt_raw < MTILES ? mt_raw : MTILES - 1;  // clamp: EXEC stays full
  const int m0 = mt << 4;
  const int n0 = blockIdx.y * LDN;

  // Cooperative staging map: thread -> (row, float4-col) of the B chunk.
  const int st_c4 = tid % ROW_F4;
  const int st_r0 = tid / ROW_F4;
  constexpr int ROWS_PER_PASS = 256 / ROW_F4;

  v8f acc[NSUB];
#pragma unroll
  for (int j = 0; j < NSUB; ++j)
    acc[j] = (v8f){0.f, 0.f, 0.f, 0.f, 0.f, 0.f, 0.f, 0.f};

  // A fragment base: lane holds A[m0+lm][k + 2*hf .. +1] (8B aligned pair)
  const float* Arow = A + (size_t)(m0 + lm) * K + (hf << 1);

  const int nchunks = K / KC;

  // stage(buf, kc0): pull B[kc0 .. kc0+KC)[n0 .. n0+LDN) into bsm[buf]
  auto stage = [&](int buf, int kc0) {
    const float* gB = B + (size_t)kc0 * N + n0;
#pragma unroll
    for (int p = 0; p < PASSES; ++p) {
      const int r = st_r0 + p * ROWS_PER_PASS;
      stage_b128(gB + (size_t)r * N + st_c4 * 4,
                 &bsm[buf][r * LDN + st_c4 * 4]);
    }
  };

  stage(0, 0);
  for (int c = 0; c < nchunks; ++c) {
    if (c + 1 < nchunks) stage((c + 1) & 1, (c + 1) * KC);
#if USE_ASYNC_LDS
    if (c + 1 < nchunks)
      __builtin_amdgcn_s_wait_asynccnt(PASSES);  // only chunk c complete
    else
      __builtin_amdgcn_s_wait_asynccnt(0);
#endif
    __syncthreads();  // bsm[c&1] visible to all waves

    // Batch-prefetch this chunk's A fragments into registers (deep MLP).
    v2f areg[KC / 4];
#pragma unroll
    for (int s = 0; s < KC / 4; ++s)
      areg[s] = *(const v2f*)(Arow + c * KC + 4 * s);

    const float* bl0 = &bsm[c & 1][(hf << 1) * LDN + lm];
#pragma unroll
    for (int s = 0; s < KC / 4; ++s) {
      const float* bl = bl0 + 4 * s * LDN;
#pragma unroll
      for (int j = 0; j < NSUB; ++j) {
        v2f b;
        b.x = bl[16 * j];        // B[k + 2*hf    ][n0 + 16j + lm]
        b.y = bl[16 * j + LDN];  // B[k + 2*hf + 1][same col]
        acc[j] = __builtin_amdgcn_wmma_f32_16x16x4_f32(
            false, areg[s], false, b, (short)0, acc[j], false, false);
      }
    }
    __syncthreads();  // everyone done reading bsm[c&1] before re-stage
  }

  if (mt_raw < MTILES) {
    // C/D layout: VGPR r -> row M = r + 8*hf, col = lane (ISA 7.12.2)
#pragma unroll
    for (int j = 0; j < NSUB; ++j) {
      float* Crow = C + (size_t)(m0 + (hf << 3)) * N + n0 + j * 16 + lm;
#pragma unroll
      for (int r = 0; r < 8; ++r) Crow[(size_t)r * N] = acc[j][r];
    }
  }
}

// ---------------------------------------------------------------------------
// Edge scatter: agg[dst] += w * support[src], 4 features per thread.
// total = E * (F/4); shift = log2(F/4).
// ---------------------------------------------------------------------------
__global__ void scatter_edges(const float* __restrict__ sup,
                              const int* __restrict__ esrc,
                              const int* __restrict__ edst,
                              const float* __restrict__ ew,
                              float* __restrict__ agg,
                              int total, int shift, int F) {
  int t = blockIdx.x * blockDim.x + threadIdx.x;
  if (t >= total) return;
  int e = t >> shift;
  int c = (t & ((1 << shift) - 1)) << 2;
  int s = esrc[e];
  int d = edst[e];
  float w = ew[e];
  const float4 v = *(const float4*)(sup + (size_t)s * F + c);
  float* o = agg + (size_t)d * F + c;
  atomicAdd(o + 0, v.x * w);
  atomicAdd(o + 1, v.y * w);
  atomicAdd(o + 2, v.z * w);
  atomicAdd(o + 3, v.w * w);
}

// ---------------------------------------------------------------------------
__global__ void zero4(float4* __restrict__ p, int n4) {
  int i = blockIdx.x * blockDim.x + threadIdx.x;
  if (i < n4) p[i] = make_float4(0.f, 0.f, 0.f, 0.f);
}

// ---------------------------------------------------------------------------
__global__ void bias_relu(float* __restrict__ h, const float* __restrict__ b,
                          int total, int fmask) {
  int i = blockIdx.x * blockDim.x + threadIdx.x;
  if (i >= total) return;
  float v = h[i] + b[i & fmask];
  h[i] = v > 0.f ? v : 0.f;
}

// ---------------------------------------------------------------------------
__global__ void bias_logsoftmax16(const float* __restrict__ agg,
                                  const float* __restrict__ bias,
                                  float* __restrict__ out, int n) {
  int row = blockIdx.x * blockDim.x + threadIdx.x;
  if (row >= n) return;
  float v[16];
  const float* p = agg + (size_t)row * 16;
#pragma unroll
  for (int i = 0; i < 16; i += 4) {
    float4 a = *(const float4*)(p + i);
    float4 bb = *(const float4*)(bias + i);
    v[i + 0] = a.x + bb.x;
    v[i + 1] = a.y + bb.y;
    v[i + 2] = a.z + bb.z;
    v[i + 3] = a.w + bb.w;
  }
  float m = v[0];
#pragma unroll
  for (int i = 1; i < 16; ++i) m = fmaxf(m, v[i]);
  float s = 0.f;
#pragma unroll
  for (int i = 0; i < 16; ++i) s += expf(v[i] - m);
  float lse = logf(s) + m;
  float* o = out + (size_t)row * 16;
#pragma unroll
  for (int i = 0; i < 16; ++i) o[i] = v[i] - lse;
}

// ---------------------------------------------------------------------------
extern "C" void kernel_launch(void* const* d_in, const int* in_sizes, int n_in,
                              void* d_out, int out_size, void* d_ws,
                              size_t ws_size, hipStream_t stream) {
  const float* x = (const float*)d_in[0];
  const int* esrc = (const int*)d_in[1];
  const int* edst = (const int*)d_in[2];
  const float* ew = (const float*)d_in[3];
  const float* W1 = (const float*)d_in[4];
  const float* b1 = (const float*)d_in[5];
  const float* W2 = (const float*)d_in[6];
  const float* b2 = (const float*)d_in[7];
  const float* W3 = (const float*)d_in[8];
  const float* b3 = (const float*)d_in[9];
  float* out = (float*)d_out;

  // Workspace layout (floats), peak = 30000*1024 floats ~ 123 MB:
  //   [0,     512N) : s1, later s2 in [0, 256N) and a2 in [256N, 512N)
  //   [512N, 1024N) : a1/h1, later s3 at [512N, +16N), a3 after it
  float* ws = (float*)d_ws;
  const size_t Nn = GCN_NODES;
  float* s1 = ws;
  float* a1 = ws + Nn * 512;
  float* s2 = ws;
  float* a2 = ws + Nn * 256;
  float* s3 = ws + Nn * 512;
  float* a3 = s3 + Nn * 16;

  const int tpb = 256;
  const int gx = (MTILES + 7) / 8;  // 8 wave M-tiles per block

  // ---------------- Layer 1: [30000,256] @ [256,512] ----------------
  gemm_wmma_tiled<4, 64><<<dim3(gx, 512 / 64), tpb, 0, stream>>>(
      x, W1, s1, 256, 512);
  {
    int n4 = GCN_NODES * 512 / 4;
    zero4<<<(n4 + tpb - 1) / tpb, tpb, 0, stream>>>((float4*)a1, n4);
  }
  {
    int tot = GCN_EDGES * (512 / 4);
    scatter_edges<<<(tot + tpb - 1) / tpb, tpb, 0, stream>>>(
        s1, esrc, edst, ew, a1, tot, 7, 512);
  }
  {
    int tot = GCN_NODES * 512;
    bias_relu<<<(tot + tpb - 1) / tpb, tpb, 0, stream>>>(a1, b1, tot, 511);
  }

  // ---------------- Layer 2: [30000,512] @ [512,256] ----------------
  gemm_wmma_tiled<4, 64><<<dim3(gx, 256 / 64), tpb, 0, stream>>>(
      a1, W2, s2, 512, 256);
  {
    int n4 = GCN_NODES * 256 / 4;
    zero4<<<(n4 + tpb - 1) / tpb, tpb, 0, stream>>>((float4*)a2, n4);
  }
  {
    int tot = GCN_EDGES * (256 / 4);
    scatter_edges<<<(tot + tpb - 1) / tpb, tpb, 0, stream>>>(
        s2, esrc, edst, ew, a2, tot, 6, 256);
  }
  {
    int tot = GCN_NODES * 256;
    bias_relu<<<(tot + tpb - 1) / tpb, tpb, 0, stream>>>(a2, b2, tot, 255);
  }

  // ---------------- Layer 3: [30000,256] @ [256,16] + log_softmax ----
  gemm_wmma_tiled<1, 64><<<dim3(gx, 1), tpb, 0, stream>>>(
      a2, W3, s3, 256, 16);
  {
    int n4 = GCN_NODES * 16 / 4;
    zero4<<<(n4 + tpb - 1) / tpb, tpb, 0, stream>>>((float4*)a3, n4);
  }
  {
    int tot = GCN_EDGES * (16 / 4);
    scatter_edges<<<(tot + tpb - 1) / tpb, tpb, 0, stream>>>(
        s3, esrc, edst, ew, a3, tot, 2, 16);
  }
  bias_logsoftmax16<<<(GCN_NODES + tpb - 1) / tpb, tpb, 0, stream>>>(
      a3, b3, out, GCN_NODES);
}